// RustIoULoss1_3272765080107
// MI455X (gfx1250) — compile-verified
//
#include <hip/hip_runtime.h>

typedef __attribute__((ext_vector_type(2))) float v2f;
typedef __attribute__((ext_vector_type(4))) float v4f;
typedef __attribute__((ext_vector_type(8))) float v8f;

#define TPB 256
#define MAIN_BLOCKS 1024
#define CHUNK 8                 // floats per thread per iteration (per array)
#define WAVE_CHUNK (32 * CHUNK) // 256 floats per wave-chunk

// D = A(16x4) * ones(4x16) + C  ->  exact f32 row-sum accumulation.
// A layout is irrelevant for a total sum (any permutation sums identically).
__device__ __forceinline__ v8f wmma_acc(v2f a, v8f c) {
    v2f ones;
    ones[0] = 1.0f;
    ones[1] = 1.0f;
    return __builtin_amdgcn_wmma_f32_16x16x4_f32(
        /*neg_a=*/false, a, /*neg_b=*/false, ones,
        /*c_mod=*/(short)0, c, /*reuse_a=*/false, /*reuse_b=*/false);
}

__device__ __forceinline__ v2f vlo(v4f v) { return __builtin_shufflevector(v, v, 0, 1); }
__device__ __forceinline__ v2f vhi(v4f v) { return __builtin_shufflevector(v, v, 2, 3); }

__device__ __forceinline__ float wave_sum32(float v) {
#pragma unroll
    for (int off = 16; off > 0; off >>= 1)
        v += __shfl_xor(v, off, 32);
    return v;
}

// Process 8 contiguous floats per stream: 4 NT b128 loads, 2 pk_mul-pairs, 12 WMMAs.
__device__ __forceinline__ void chunk8(const float* __restrict__ P,
                                       const float* __restrict__ G,
                                       long long base,
                                       v8f& accP, v8f& accG, v8f& accPG) {
    const v4f p0 = __builtin_nontemporal_load((const v4f*)(P + base));
    const v4f p1 = __builtin_nontemporal_load((const v4f*)(P + base + 4));
    const v4f g0 = __builtin_nontemporal_load((const v4f*)(G + base));
    const v4f g1 = __builtin_nontemporal_load((const v4f*)(G + base + 4));
    const v4f m0 = p0 * g0;
    const v4f m1 = p1 * g1;
    accP = wmma_acc(vlo(p0), accP);
    accP = wmma_acc(vhi(p0), accP);
    accP = wmma_acc(vlo(p1), accP);
    accP = wmma_acc(vhi(p1), accP);
    accG = wmma_acc(vlo(g0), accG);
    accG = wmma_acc(vhi(g0), accG);
    accG = wmma_acc(vlo(g1), accG);
    accG = wmma_acc(vhi(g1), accG);
    accPG = wmma_acc(vlo(m0), accPG);
    accPG = wmma_acc(vhi(m0), accPG);
    accPG = wmma_acc(vlo(m1), accPG);
    accPG = wmma_acc(vhi(m1), accPG);
}

// ---------------------------------------------------------------------------
// Kernel 1: streaming triple reduction (sum P, sum G, sum P*G) via f32 WMMA
// accumulators. Deterministic: per-block partials, no float atomics.
// ---------------------------------------------------------------------------
__global__ __launch_bounds__(TPB)
void triple_reduce_kernel(const float* __restrict__ P,
                          const float* __restrict__ G,
                          float* __restrict__ part_p,
                          float* __restrict__ part_g,
                          float* __restrict__ part_pg,
                          long long N) {
    const int tid = blockIdx.x * blockDim.x + threadIdx.x;
    const long long stride = (long long)gridDim.x * blockDim.x * CHUNK;
    const long long Nmain = N & ~(long long)(WAVE_CHUNK - 1);  // wave-uniform bound

    v8f accP = {};
    v8f accG = {};
    v8f accPG = {};

    // Scalar (wave-uniform) trip count: no per-iteration EXEC masking, and
    // EXEC is trivially all-1s around every WMMA.
    long long base = (long long)tid * CHUNK;
    const long long nfull = Nmain / stride;
    for (long long it = 0; it < nfull; ++it) {
        chunk8(P, G, base, accP, accG, accPG);
        base += stride;
    }
    if (base < Nmain) {  // remainder chunk; uniform across each wave
        chunk8(P, G, base, accP, accG, accPG);
    }

    // Scalar tail (N not a multiple of WAVE_CHUNK) — block 0, no WMMA.
    float tp = 0.0f, tg = 0.0f, tpg = 0.0f;
    if (blockIdx.x == 0) {
        const long long rem = N - Nmain;  // < 256
        if ((long long)threadIdx.x < rem) {
            const float p = P[Nmain + threadIdx.x];
            const float g = G[Nmain + threadIdx.x];
            tp = p; tg = g; tpg = p * g;
        }
    }

    // Fold the 8 C/D VGPRs per lane. Summing all 32 lanes then yields 16x the
    // true total (16 identical columns) -> scale by 1/16 (exact, pow2).
    float sp = 0.0f, sg = 0.0f, spg = 0.0f;
#pragma unroll
    for (int r = 0; r < 8; ++r) {
        sp += accP[r];
        sg += accG[r];
        spg += accPG[r];
    }
    sp  += tp  * 16.0f;  // pre-scale tail so the 1/16 below cancels exactly
    sg  += tg  * 16.0f;
    spg += tpg * 16.0f;

    sp  = wave_sum32(sp)  * 0.0625f;
    sg  = wave_sum32(sg)  * 0.0625f;
    spg = wave_sum32(spg) * 0.0625f;

    __shared__ float lds[TPB / 32][3];
    const int wave = threadIdx.x >> 5;
    const int lane = threadIdx.x & 31;
    if (lane == 0) {
        lds[wave][0] = sp;
        lds[wave][1] = sg;
        lds[wave][2] = spg;
    }
    __syncthreads();
    if (threadIdx.x == 0) {
        float bp = 0.0f, bg = 0.0f, bpg = 0.0f;
#pragma unroll
        for (int w = 0; w < TPB / 32; ++w) {  // fixed order -> deterministic
            bp += lds[w][0];
            bg += lds[w][1];
            bpg += lds[w][2];
        }
        part_p[blockIdx.x] = bp;
        part_g[blockIdx.x] = bg;
        part_pg[blockIdx.x] = bpg;
    }
}

// ---------------------------------------------------------------------------
// Kernel 2: one block per (b, k) region; replicates _bounds() clipping.
// ---------------------------------------------------------------------------
__device__ __forceinline__ void region_bounds(int c, int dim, int& s, int& e) {
    s = max(c - 20, 0);
    e = min(c + 20, dim);
    const int nw = e - s;
    const bool odd_small = ((nw & 1) != 0) && (nw < 40);
    if (odd_small && s == 0) e -= 1;
    if (odd_small && e == dim) s += 1;  // uses updated e, matching jnp order
}

__global__ __launch_bounds__(TPB)
void region_iou_kernel(const float* __restrict__ P,
                       const float* __restrict__ G,
                       const int* __restrict__ cents,
                       float* __restrict__ region_iou,
                       int K, int H, int W) {
    const int b = blockIdx.x / K;
    const int k = blockIdx.x % K;
    const int cy = cents[(b * K + k) * 2 + 0];
    const int cx = cents[(b * K + k) * 2 + 1];

    int sy, ey, sx, ex;
    region_bounds(cy, H, sy, ey);
    region_bounds(cx, W, sx, ex);

    const int ny = ey - sy;
    const int nx = ex - sx;
    const int total = ny * nx;
    const float* __restrict__ Pb = P + (size_t)b * H * W;
    const float* __restrict__ Gb = G + (size_t)b * H * W;

    float inter = 0.0f, ps = 0.0f, gs = 0.0f;
    for (int i = threadIdx.x; i < total; i += blockDim.x) {
        const int yy = sy + i / nx;
        const int xx = sx + i % nx;
        const float p = Pb[(size_t)yy * W + xx];
        const float g = Gb[(size_t)yy * W + xx];
        inter += p * g;
        ps += p;
        gs += g;
    }

    __shared__ float s0[TPB], s1[TPB], s2[TPB];
    const int t = threadIdx.x;
    s0[t] = inter; s1[t] = ps; s2[t] = gs;
    __syncthreads();
#pragma unroll
    for (int off = TPB / 2; off > 0; off >>= 1) {
        if (t < off) {
            s0[t] += s0[t + off];
            s1[t] += s1[t + off];
            s2[t] += s2[t + off];
        }
        __syncthreads();
    }
    if (t == 0) {
        const float it = s0[0], pp = s1[0], gg = s2[0];
        region_iou[blockIdx.x] = (it + 1.0f) / (pp + gg - it + 1.0f);
    }
}

// ---------------------------------------------------------------------------
// Kernel 3: deterministic final combine -> d_out[0].
// ---------------------------------------------------------------------------
__global__ __launch_bounds__(TPB)
void final_kernel(const float* __restrict__ part_p,
                  const float* __restrict__ part_g,
                  const float* __restrict__ part_pg,
                  const float* __restrict__ region_iou,
                  float* __restrict__ out,
                  int nparts, int nregions) {
    __shared__ float s0[TPB], s1[TPB], s2[TPB], s3[TPB];
    const int t = threadIdx.x;
    float sp = 0.0f, sg = 0.0f, spg = 0.0f, si = 0.0f;
    for (int i = t; i < nparts; i += TPB) {
        sp += part_p[i];
        sg += part_g[i];
        spg += part_pg[i];
    }
    for (int i = t; i < nregions; i += TPB) si += region_iou[i];
    s0[t] = sp; s1[t] = sg; s2[t] = spg; s3[t] = si;
    __syncthreads();
#pragma unroll
    for (int off = TPB / 2; off > 0; off >>= 1) {
        if (t < off) {
            s0[t] += s0[t + off];
            s1[t] += s1[t + off];
            s2[t] += s2[t + off];
            s3[t] += s3[t + off];
        }
        __syncthreads();
    }
    if (t == 0) {
        const float psum = s0[0], gsum = s1[0], inter = s2[0];
        const float loss_global = 1.0f - (inter + 1.0f) / (psum + gsum - inter + 1.0f);
        const float loss_region = 1.0f - s3[0] / (float)nregions;
        out[0] = loss_global + loss_region;
    }
}

// ---------------------------------------------------------------------------
extern "C" void kernel_launch(void* const* d_in, const int* in_sizes, int n_in,
                              void* d_out, int out_size, void* d_ws, size_t ws_size,
                              hipStream_t stream) {
    const float* P = (const float*)d_in[0];       // preds  (B,1,H,W) f32
    const float* G = (const float*)d_in[1];       // gt     (B,1,H,W) f32
    const int* C = (const int*)d_in[2];           // cents  (B,K,2) i32

    const long long N = (long long)in_sizes[0];
    const int H = 1024, W = 1024;                 // reference setup
    const int B = (int)(N / ((long long)H * W));
    const int K = in_sizes[2] / (2 * B);
    const int nregions = B * K;

    float* ws = (float*)d_ws;
    float* part_p = ws;                           // [MAIN_BLOCKS]
    float* part_g = ws + MAIN_BLOCKS;             // [MAIN_BLOCKS]
    float* part_pg = ws + 2 * MAIN_BLOCKS;        // [MAIN_BLOCKS]
    float* region_iou = ws + 3 * MAIN_BLOCKS;     // [nregions]

    triple_reduce_kernel<<<MAIN_BLOCKS, TPB, 0, stream>>>(P, G, part_p, part_g, part_pg, N);
    region_iou_kernel<<<nregions, TPB, 0, stream>>>(P, G, C, region_iou, K, H, W);
    final_kernel<<<1, TPB, 0, stream>>>(part_p, part_g, part_pg, region_iou,
                                        (float*)d_out, MAIN_BLOCKS, nregions);
}